// Normalization_60095182406123
// MI455X (gfx1250) — compile-verified
//
#include <hip/hip_runtime.h>
#include <hip/hip_bf16.h>
#include <math.h>

// ---------------------------------------------------------------------------
// Pipeline:
//  Stage 1 (WMMA, fused H+V 32-tap Gaussian blur on squared+padded planes):
//    768 planes of 224x224; one workgroup = one 16-row x 224-col output strip.
//    Both 1-D convs are band-matrix GEMMs on V_WMMA_F32_16X16X4_F32
//    (K padded 47->48 = 12 K=4 chunks per 16x16 tile). fp32 precision kept.
//    Gaussian band coefficients are built ONCE into a zero-guarded LDS table
//    (32 expf total per block) instead of 24 expf chains per thread.
//    Loader: pure b128 global loads (every 4-col quad of the padded window is
//    fully interior or fully zero since PL=16, SZ=224 are multiples of 4).
//  Stage 2: 3x3 separable Gaussian over the 12x8 (freq,orient) grid per pixel
//    + final [img,freq,orient,phase,H,W] permutation. Pure bandwidth.
// Traffic floor ~617 MB -> ~26 us at 23.3 TB/s; WMMA keeps compute far under.
// ---------------------------------------------------------------------------

typedef float v2f __attribute__((ext_vector_type(2)));
typedef float v8f __attribute__((ext_vector_type(8)));

#define NIMG   4
#define NFEAT  192
#define SZ     224
#define NPLANE (NIMG * NFEAT)      // 768
#define PLANE_PIX (SZ * SZ)        // 50176
#define NFREQ  12
#define NORIENT 8
#define NPHASE 2

// LDS strides (floats).
// SIN_STRIDE 260: mult of 4 -> 16B-aligned rows for b128 stores; 260%64=4 ->
//   A-loads: lanes0-15 banks 4L (distinct), lanes16-31 banks 4L+2 (disjoint).
// TMP_STRIDE 232: 2*232 % 64 = 16 -> vertical B-loads conflict-free halves.
#define SIN_STRIDE 260
#define TMP_STRIDE 232
#define NROWBLK 14                 // 224/16 row strips per plane
#define NWAVES  14                 // 224/16 column slabs, one wave each
#define BLKTHREADS (NWAVES * 32)   // 448

__global__ __launch_bounds__(BLKTHREADS) void
stage1_blur_wmma(const float* __restrict__ x, float* __restrict__ mid) {
  __shared__ float sIn[48 * SIN_STRIDE];   // squared, padded input window
  __shared__ float sTmp[48 * TMP_STRIDE];  // after horizontal pass
  __shared__ float gtab[64];               // zero-guarded gauss table: gtab[16+i]=g[i]

  const int plane = blockIdx.x / NROWBLK;  // img*192 + feat (plane-major input)
  const int blk   = blockIdx.x % NROWBLK;  // 16-row output strip index
  const float* __restrict__ src = x   + (size_t)plane * PLANE_PIX;
  float*       __restrict__ dst = mid + (size_t)plane * PLANE_PIX;

  const int tid = threadIdx.x;
  const int pr0 = blk * 16;                // first padded row of 48-row window

  // ---- gauss table: g[i] = exp(-u^2/2)/sqrt(2pi), u = -1 + 2i/31 ----
  if (tid < 64) gtab[tid] = 0.0f;
  // (zero guards cover band indices outside [0,32))

  // ---- cooperative load: square + zero-pad (PT=PL=16, PB=PR=15) ----
  // 48 rows x 64 quads; quad is fully interior iff 16 <= pc4 <= 236 & row ok.
  for (int idx = tid; idx < 48 * 64; idx += BLKTHREADS) {
    const int rr  = idx >> 6;              // 0..47
    const int pc4 = (idx & 63) * 4;        // padded col quad start (0..252)
    const int r   = pr0 + rr - 16;         // original row
    float4 v = make_float4(0.f, 0.f, 0.f, 0.f);
    if (r >= 0 && r < SZ && pc4 >= 16 && pc4 <= 236) {
      const float4 t = *(const float4*)&src[r * SZ + (pc4 - 16)];
      v = make_float4(t.x * t.x, t.y * t.y, t.z * t.z, t.w * t.w);
    }
    *(float4*)&sIn[rr * SIN_STRIDE + pc4] = v;
  }
  if (tid < 32) {
    const float u = -1.0f + (2.0f / 31.0f) * (float)tid;
    gtab[16 + tid] = 0.3989422804f * expf(-0.5f * u * u);
  }
  __syncthreads();

  const int wave = tid >> 5;               // 0..13 -> 16-col slab
  const int lane = tid & 31;
  const int cs   = wave * 16;
  const int m    = lane & 15;              // M (A/C) or N (B/C) index
  const int hi   = lane >> 4;              // half-wave select

  // Band coefficients, identical per lane for horizontal-B and vertical-A:
  //   G[k][j] = g[k - j], k = 4q + 2*hi + {0,1}, j = m.
  float g0[12], g1[12];
#pragma unroll
  for (int q = 0; q < 12; ++q) {
    const int b = 16 + 4 * q + 2 * hi - m; // 1..62
    g0[q] = gtab[b];
    g1[q] = gtab[b + 1];
  }

  // ---- horizontal pass: tmp[r][cs+c] = sum_k sIn[r][cs+k] * G[k][c] ----
#pragma unroll
  for (int rg = 0; rg < 3; ++rg) {         // 3 x 16 rows = 48 tmp rows
    const int rbase = rg * 16;
    v8f acc = {};
#pragma unroll
    for (int q = 0; q < 12; ++q) {
      // A: data, lane layout (M=m, K=4q+2*hi+{0,1}) -> aligned ds b64 load.
      const v2f a = *(const v2f*)&sIn[(rbase + m) * SIN_STRIDE + cs + 4 * q + 2 * hi];
      const v2f b = { g0[q], g1[q] };      // B: band coefficients
      acc = __builtin_amdgcn_wmma_f32_16x16x4_f32(
          false, a, false, b, (short)0, acc, false, false);
    }
#pragma unroll
    for (int v = 0; v < 8; ++v)            // C/D: M = v + 8*hi, N = m
      sTmp[(rbase + v + 8 * hi) * TMP_STRIDE + cs + m] = acc[v];
  }
  __syncthreads();

  // ---- vertical pass: out[ro][c] = sum_k G[k][ro] * tmp[k][c] ----
  v8f acc = {};
#pragma unroll
  for (int q = 0; q < 12; ++q) {
    const v2f a = { g0[q], g1[q] };        // A: band coefficients (transposed)
    const int r0 = 4 * q + 2 * hi;         // B: data (K=r0+{0,1}, N=m)
    const v2f b = { sTmp[r0 * TMP_STRIDE + cs + m],
                    sTmp[(r0 + 1) * TMP_STRIDE + cs + m] };
    acc = __builtin_amdgcn_wmma_f32_16x16x4_f32(
        false, a, false, b, (short)0, acc, false, false);
  }
#pragma unroll
  for (int v = 0; v < 8; ++v) {
    const int orow = blk * 16 + v + 8 * hi;
    dst[orow * SZ + cs + m] = acc[v];      // coalesced along lanes (cols)
  }
}

// ---------------------------------------------------------------------------
// Stage 2: per (img, phase, pixel): 3x3 separable gaussian over (freq, orient)
// with zero padding of 1; writes [img][freq][orient][phase][H][W].
// g3 = [exp(-.5), 1, exp(-.5)] / sqrt(2*pi)
// ---------------------------------------------------------------------------
__global__ __launch_bounds__(256) void
stage2_fo_stencil(const float* __restrict__ mid, float* __restrict__ out) {
  const int total = NIMG * NPHASE * PLANE_PIX;     // 401408
  const int t = blockIdx.x * blockDim.x + threadIdx.x;
  if (t >= total) return;

  const int hw = t % PLANE_PIX;                    // pixel (h*224+w)
  const int ph = (t / PLANE_PIX) % NPHASE;
  const int im = t / (PLANE_PIX * NPHASE);

  const float gc = 0.3989422804f;                  // g3[1]
  const float ge = 0.2419707245f;                  // g3[0] = g3[2]

  // mid channel index: feat = (f*8 + o)*2 + ph, layout [img][feat][H][W]
  const float* __restrict__ base =
      mid + ((size_t)im * NFEAT + ph) * PLANE_PIX + hw;
  float* __restrict__ obase =
      out + (((size_t)im * NFREQ * NORIENT * NPHASE) + ph) * PLANE_PIX + hw;

  float rm1[NORIENT], r0[NORIENT], rp1[NORIENT];
#pragma unroll
  for (int o = 0; o < NORIENT; ++o) {
    rm1[o] = 0.0f;
    r0[o]  = base[(size_t)(o * NPHASE) * PLANE_PIX];          // f = 0 row
  }

#pragma unroll
  for (int f = 0; f < NFREQ; ++f) {
#pragma unroll
    for (int o = 0; o < NORIENT; ++o)
      rp1[o] = (f + 1 < NFREQ)
                 ? base[(size_t)(((f + 1) * NORIENT + o) * NPHASE) * PLANE_PIX]
                 : 0.0f;

    float vf[NORIENT];
#pragma unroll
    for (int o = 0; o < NORIENT; ++o)                // freq-direction blend
      vf[o] = ge * (rm1[o] + rp1[o]) + gc * r0[o];

#pragma unroll
    for (int o = 0; o < NORIENT; ++o) {              // orient-direction blend
      const float left  = (o > 0)           ? vf[o - 1] : 0.0f;
      const float right = (o < NORIENT - 1) ? vf[o + 1] : 0.0f;
      const float val   = gc * vf[o] + ge * (left + right);
      obase[(size_t)((f * NORIENT + o) * NPHASE) * PLANE_PIX] = val;
    }
#pragma unroll
    for (int o = 0; o < NORIENT; ++o) { rm1[o] = r0[o]; r0[o] = rp1[o]; }
  }
}

extern "C" void kernel_launch(void* const* d_in, const int* in_sizes, int n_in,
                              void* d_out, int out_size, void* d_ws, size_t ws_size,
                              hipStream_t stream) {
  const float* x = (const float*)d_in[0];
  float* out = (float*)d_out;
  float* mid = (float*)d_ws;   // 768*224*224 floats (~154 MB) intermediate

  stage1_blur_wmma<<<NPLANE * NROWBLK, BLKTHREADS, 0, stream>>>(x, mid);

  const int total2 = NIMG * NPHASE * PLANE_PIX;
  stage2_fo_stencil<<<(total2 + 255) / 256, 256, 0, stream>>>(mid, out);
}